// FeatureWiseEmbedding_72292889526695
// MI455X (gfx1250) — compile-verified
//
#include <hip/hip_runtime.h>
#include <hip/hip_bf16.h>

// Problem constants (from the reference)
#define B_SZ   16384
#define F_SZ   100
#define NCAT   28
#define CARD   100
#define C_SZ   5
#define DTOK   128
#define NTOK   (NCAT + F_SZ)   // 128

typedef __attribute__((ext_vector_type(2))) float v2f;
typedef __attribute__((ext_vector_type(4))) float v4f;
typedef __attribute__((ext_vector_type(8))) float v8f;

// ---------------------------------------------------------------------------
// Kernel 1: numeric tokens via V_WMMA_F32_16X16X4_F32
//   out[b, NCAT+f, d] = sum_c PReLU(x_num[b,f]*W[f,c]+Bias[f,c]) * TW[f,c,d]
//                       + tok_bias[NCAT+f, d]
// One wave handles a 16(batch) x 128(d) output tile for one feature f.
// K = C = 5 is padded to 8, split into two K=4 WMMAs. The padded K slots
// (5,6,7) are killed on the A side only (masked once, outside the d-loop),
// so all B-side loads are unconditional -> no divergent branches.
// ---------------------------------------------------------------------------
__global__ __launch_bounds__(256) void
num_tok_wmma_kernel(const float* __restrict__ x_num,
                    const float* __restrict__ fwe_w,
                    const float* __restrict__ fwe_b,
                    const float* __restrict__ prelu_a,
                    const float* __restrict__ tok_w,
                    const float* __restrict__ tok_bias,
                    float* __restrict__ out)
{
    const int lane = threadIdx.x & 31;
    const int wsub = threadIdx.x >> 5;           // wave index in block (0..7)
    const int w    = blockIdx.x * 8 + wsub;      // global wave id
    const int f    = w % F_SZ;                   // feature
    const int b0   = (w / F_SZ) * 16;            // batch tile start

    const int m  = lane & 15;                    // A row / C-D column index
    const int kh = lane >> 4;                    // lane-half (K offset selector)

    const float alpha = prelu_a[0];
    const float x = x_num[(size_t)(b0 + m) * F_SZ + f];

    // --- A matrix (16x4 f32 layout: VGPR j holds K = 2*kh + j) --------------
    // Chunk 0 covers K/c = 0..3 (all real). Chunk 1 covers c = 4..7 (only
    // c==4 real; it lives in slot j=0 of the kh==0 lane half).
    v2f a0, a1;
#pragma unroll
    for (int j = 0; j < 2; ++j) {
        const int c = 2 * kh + j;                // 0..3, always valid
        const float v = x * fwe_w[f * C_SZ + c] + fwe_b[f * C_SZ + c];
        a0[j] = (v >= 0.0f) ? v : alpha * v;
    }
    {
        // h at c = 4, computed by all lanes (valid address), masked to kh==0.
        const float v  = x * fwe_w[f * C_SZ + 4] + fwe_b[f * C_SZ + 4];
        const float h4 = (v >= 0.0f) ? v : alpha * v;
        a1[0] = (kh == 0) ? h4 : 0.0f;           // K=4 real, K=6 padded
        a1[1] = 0.0f;                            // K=5 / K=7 padded
    }

    const int n = m;                                           // d column in tile
    const float* twp = tok_w + (size_t)f * C_SZ * DTOK + n;    // tok_w[f, 0, n]
    const float* tbp = tok_bias + (size_t)(NCAT + f) * DTOK + n;
    float* outp = out + ((size_t)(b0 + kh * 8) * NTOK + NCAT + f) * DTOK + n;

#pragma unroll
    for (int dt = 0; dt < 8; ++dt) {
        const int d0 = dt * 16;

        // --- B matrix (4x16 f32 layout: VGPR j holds row K = 2*kh + j) ------
        v2f bb0, bb1;
#pragma unroll
        for (int j = 0; j < 2; ++j) {
            const int c = 2 * kh + j;            // 0..3, always valid
            bb0[j] = twp[(size_t)c * DTOK + d0];
        }
        // K=4 row, loaded by ALL lanes (kh==1 lanes pair with a1[0]==0).
        bb1[0] = twp[(size_t)4 * DTOK + d0];
        bb1[1] = 0.0f;                           // finite pad (avoid 0*garbage)

        v8f acc = {};
        acc = __builtin_amdgcn_wmma_f32_16x16x4_f32(false, a0, false, bb0,
                                                    (short)0, acc, false, false);
        acc = __builtin_amdgcn_wmma_f32_16x16x4_f32(false, a1, false, bb1,
                                                    (short)0, acc, false, false);

        const float tb = tbp[d0];

        // C/D layout: VGPR r -> row (r + 8*kh), col n. Streaming stores.
#pragma unroll
        for (int r = 0; r < 8; ++r) {
            __builtin_nontemporal_store(acc[r] + tb,
                                        outp + (size_t)r * NTOK * DTOK + d0);
        }
    }
}

// ---------------------------------------------------------------------------
// Kernel 2: category token gather
//   out[b, j, d] = cat_emb[x_cat[b,j] + offsets[j], d] + tok_bias[j, d]
// One thread per float4 of output; exact grid, no tail.
// ---------------------------------------------------------------------------
__global__ __launch_bounds__(256) void
cat_tok_kernel(const int* __restrict__ x_cat,
               const int* __restrict__ cat_offsets,
               const float* __restrict__ cat_emb,
               const float* __restrict__ tok_bias,
               float* __restrict__ out)
{
    const size_t gid = (size_t)blockIdx.x * 256 + threadIdx.x;
    const int q = (int)(gid & 31);                 // float4 slot within D=128
    const int j = (int)((gid >> 5) % NCAT);        // category feature
    const int b = (int)(gid / (32u * NCAT));       // batch row

    const int row = x_cat[(size_t)b * NCAT + j] + cat_offsets[j];

    const v4f e  = ((const v4f*)cat_emb)[(size_t)row * 32 + q];
    const v4f tb = ((const v4f*)tok_bias)[(size_t)j * 32 + q];
    const v4f o  = e + tb;

    v4f* p = (v4f*)out + ((size_t)b * NTOK + j) * 32 + q;
    __builtin_nontemporal_store(o, p);
}

// ---------------------------------------------------------------------------
extern "C" void kernel_launch(void* const* d_in, const int* in_sizes, int n_in,
                              void* d_out, int out_size, void* d_ws, size_t ws_size,
                              hipStream_t stream)
{
    const float* x_num    = (const float*)d_in[0];
    const int*   x_cat    = (const int*)  d_in[1];
    const float* fwe_w    = (const float*)d_in[2];
    const float* fwe_b    = (const float*)d_in[3];
    const float* prelu_a  = (const float*)d_in[4];
    const float* tok_w    = (const float*)d_in[5];
    const float* tok_bias = (const float*)d_in[6];
    const float* cat_emb  = (const float*)d_in[7];
    const int*   offsets  = (const int*)  d_in[8];
    float* out = (float*)d_out;

    // Numeric tokens: (B/16) * F waves, 8 waves per 256-thread block.
    constexpr int num_waves  = (B_SZ / 16) * F_SZ;        // 102400
    constexpr int num_blocks = num_waves / 8;             // 12800 (exact)
    num_tok_wmma_kernel<<<num_blocks, 256, 0, stream>>>(
        x_num, fwe_w, fwe_b, prelu_a, tok_w, tok_bias, out);

    // Category tokens: one thread per float4, exact multiple of 256.
    constexpr int cat_threads = B_SZ * NCAT * (DTOK / 4); // 14,680,064
    constexpr int cat_blocks  = cat_threads / 256;        // 57,344 (exact)
    cat_tok_kernel<<<cat_blocks, 256, 0, stream>>>(
        x_cat, offsets, cat_emb, tok_bias, out);
}